// TensorTrainEmbedding_24077586661461
// MI455X (gfx1250) — compile-verified
//
#include <hip/hip_runtime.h>
#include <stdint.h>

#ifndef __has_builtin
#define __has_builtin(x) 0
#endif

#define DIM        64
#define RANK       16
#define HRANGE     8192
#define ROW_BYTES  1024u                    // RANK*RANK*4 bytes per d
#define PAD_BYTES  64u                      // per-row LDS pad (bank-conflict kill)
#define ROWP_BYTES (ROW_BYTES + PAD_BYTES)  // 1088
#define SLICE_FLOATS (DIM * RANK * RANK)    // 16384 floats = 64 KB
#define SLICEP_BYTES (DIM * ROWP_BYTES)     // 69632 bytes in LDS
#define VSTRIDE    20                       // padded v row stride (floats)

typedef int   v4i __attribute__((ext_vector_type(4)));
typedef float v2f __attribute__((ext_vector_type(2)));
typedef float v8f __attribute__((ext_vector_type(8)));
typedef __attribute__((address_space(1))) v4i gv4i_t;   // global int4
typedef __attribute__((address_space(3))) v4i lv4i_t;   // LDS int4

// Async DMA: 16 bytes/lane, global -> LDS, tracked by ASYNCcnt (CDNA5 path).
__device__ __forceinline__ void async_b128(const void* gsrc, void* ldst) {
#if __has_builtin(__builtin_amdgcn_global_load_async_to_lds_b128)
  __builtin_amdgcn_global_load_async_to_lds_b128(
      (gv4i_t*)(uintptr_t)gsrc,
      (lv4i_t*)(uint32_t)(uintptr_t)ldst,   // low 32 bits of generic = LDS addr
      0, 0);
#else
  uint32_t laddr = (uint32_t)(uintptr_t)ldst;
  asm volatile("global_load_async_to_lds_b128 %0, %1, off"
               :: "v"(laddr), "v"(gsrc)
               : "memory");
#endif
}

__device__ __forceinline__ void wait_async_16() {
#if __has_builtin(__builtin_amdgcn_s_wait_asynccnt)
  __builtin_amdgcn_s_wait_asynccnt(16);
#else
  asm volatile("s_wait_asynccnt 16" ::: "memory");
#endif
}

__device__ __forceinline__ void wait_async_0() {
#if __has_builtin(__builtin_amdgcn_s_wait_asynccnt)
  __builtin_amdgcn_s_wait_asynccnt(0);
#else
  asm volatile("s_wait_asynccnt 0" ::: "memory");
#endif
}

__global__ void __launch_bounds__(256)
tt_embed_kernel(const int*   __restrict__ hs,
                const float* __restrict__ start_core,
                const float* __restrict__ end_core,
                const float* __restrict__ cores,
                float*       __restrict__ out)
{
  __shared__ __align__(16) char  lds_m[2][SLICEP_BYTES];  // double-buffered core slices
  __shared__ float               lds_v[DIM * VSTRIDE];    // running v[d][r], padded

  const uint32_t b = blockIdx.x;
  const uint32_t t = threadIdx.x;     // 0..255
  const uint32_t d4 = t >> 2;         // per-thread d for E/S handling
  const uint32_t q  = t & 3;

  const int h0 = hs[4*b + 0];
  const int h1 = hs[4*b + 1];
  const int h2 = hs[4*b + 2];
  const int h3 = hs[4*b + 3];

  const char* c0 = (const char*)(cores + (size_t)h1 * SLICE_FLOATS);
  const char* c1 = (const char*)(cores + ((size_t)HRANGE + (size_t)h2) * SLICE_FLOATS);

  // Kick off both 64 KB slice copies ASAP; slice0 completes first (in-order).
  // LDS target inserts 64 B pad per 1 KB d-row: loff = goff + (goff>>10)*64.
  #pragma unroll
  for (int k = 0; k < 16; ++k) {
    uint32_t goff = t * 16u + (uint32_t)k * 4096u;
    uint32_t loff = goff + ((goff >> 10) << 6);
    async_b128(c0 + goff, &lds_m[0][loff]);
  }
  #pragma unroll
  for (int k = 0; k < 16; ++k) {
    uint32_t goff = t * 16u + (uint32_t)k * 4096u;
    uint32_t loff = goff + ((goff >> 10) << 6);
    async_b128(c1 + goff, &lds_m[1][loff]);
  }

  // v0 = end_core[h3] : coalesced float4 per lane -> padded LDS rows
  {
    const float4 e = *(const float4*)(end_core + (size_t)h3 * (DIM*RANK) + t * 4u);
    float* vd = &lds_v[d4 * VSTRIDE + q * 4u];
    vd[0] = e.x; vd[1] = e.y; vd[2] = e.z; vd[3] = e.w;
  }

  // Wave-cooperative mapping for the WMMA mid-steps
  const uint32_t wave = t >> 5;       // 0..7
  const uint32_t lane = t & 31;
  const uint32_t hi   = lane >> 4;    // lane half
  const uint32_t n    = lane & 15;    // column index (s)

#if __has_builtin(__builtin_amdgcn_wmma_f32_16x16x4_f32)
  // ---- mid steps on the matrix pipe ----
  // A[m,k] = v[4c+k] (broadcast over m), B[k,n] = M[4c+k][n]:
  // D[m,n] = sum_k v[4c+k]*M[4c+k][n] accumulated over c -> v'[n] in every D VGPR.
  // f32 operand layout (ISA 7.12.2): VGPR j, lane-half hi  <->  K = 2*hi + j.
  #pragma unroll
  for (int step = 0; step < 2; ++step) {
    if (step == 0) wait_async_16(); else wait_async_0();
    __syncthreads();                 // slice resident everywhere; v visible

    float vnew[8];
    #pragma unroll
    for (int dd = 0; dd < 8; ++dd) {
      const uint32_t d = wave * 8u + (uint32_t)dd;
      const float* vrow = &lds_v[d * VSTRIDE];
      const float* mrow = (const float*)&lds_m[step][d * ROWP_BYTES];
      v8f acc = {0.f, 0.f, 0.f, 0.f, 0.f, 0.f, 0.f, 0.f};
      #pragma unroll
      for (int c = 0; c < 4; ++c) {
        const uint32_t r0 = 4u * c + 2u * hi;       // K = 2*hi + j, j = vgpr
        v2f A, B;
        A.x = vrow[r0 + 0];
        A.y = vrow[r0 + 1];
        B.x = mrow[(r0 + 0) * 16u + n];
        B.y = mrow[(r0 + 1) * 16u + n];
        acc = __builtin_amdgcn_wmma_f32_16x16x4_f32(
            false, A, false, B, (short)0, acc, false, false);
      }
      vnew[dd] = acc[0];             // v'[n], replicated in all D rows
    }
    __syncthreads();                 // all reads of v done (WAR)
    if (hi == 0) {                   // lanes 0..15 write v'[n] per d
      #pragma unroll
      for (int dd = 0; dd < 8; ++dd)
        lds_v[(wave * 8u + (uint32_t)dd) * VSTRIDE + n] = vnew[dd];
    }
    __syncthreads();                 // v' visible
  }
#else
  // ---- fallback: per-thread VALU FMA mid-steps ----
  #pragma unroll
  for (int step = 0; step < 2; ++step) {
    if (step == 0) wait_async_16(); else wait_async_0();
    __syncthreads();
    const char*  mrow = &lds_m[step][d4 * ROWP_BYTES + q * 16u];
    const float* v    = &lds_v[d4 * VSTRIDE];
    float a0 = 0.f, a1 = 0.f, a2 = 0.f, a3 = 0.f;
    #pragma unroll
    for (int r = 0; r < RANK; ++r) {
      const float  vr = v[r];
      const float4 m  = *(const float4*)(mrow + r * 64u);
      a0 = fmaf(vr, m.x, a0);
      a1 = fmaf(vr, m.y, a1);
      a2 = fmaf(vr, m.z, a2);
      a3 = fmaf(vr, m.w, a3);
    }
    __syncthreads();
    float* vd = &lds_v[d4 * VSTRIDE + q * 4u];
    vd[0] = a0; vd[1] = a1; vd[2] = a2; vd[3] = a3;
    __syncthreads();
  }
#endif

  // ---- final: out[b,d] = sum_r S[h0][d,r] * v2[d,r] ----
  const float4 s4 = *(const float4*)(start_core + (size_t)h0 * (DIM*RANK) + t * 4u);
  const float* v  = &lds_v[d4 * VSTRIDE + q * 4u];
  float p = s4.x * v[0] + s4.y * v[1] + s4.z * v[2] + s4.w * v[3];
  // reduce the 4 q-lanes (wave32 shuffles)
  p += __shfl_xor(p, 1, 32);
  p += __shfl_xor(p, 2, 32);
  if (q == 0) out[(size_t)b * DIM + d4] = p;
}

extern "C" void kernel_launch(void* const* d_in, const int* in_sizes, int n_in,
                              void* d_out, int out_size, void* d_ws, size_t ws_size,
                              hipStream_t stream) {
  const int*   hs         = (const int*)  d_in[0];
  const float* start_core = (const float*)d_in[1];
  const float* end_core   = (const float*)d_in[2];
  const float* cores      = (const float*)d_in[3];
  float*       out        = (float*)d_out;

  const int B = in_sizes[0] / 4;   // 16384 rows, one block per row
  tt_embed_kernel<<<B, 256, 0, stream>>>(hs, start_core, end_core, cores, out);
}